// MyModel_78537771974927
// MI455X (gfx1250) — compile-verified
//
#include <hip/hip_runtime.h>
#include <math.h>

// HGPSL pipeline for MI455X (gfx1250, wave32).
// GEMMs use V_WMMA_F32_16X16X4_F32 (fp32 matrix pipe, matches fp32 reference).
// Edge aggregation via f32 global atomics; dense pooled adjacency built by
// rank-scatter (never materializing the 512x512 adjacency).

typedef float v2f __attribute__((ext_vector_type(2)));
typedef float v8f __attribute__((ext_vector_type(8)));

#define CDIV(a, b) (((a) + (b) - 1) / (b))

// ---------------------------------------------------------------------------
// Zero fill
// ---------------------------------------------------------------------------
__global__ void zero_f32(float* __restrict__ p, size_t n) {
  size_t i = blockIdx.x * (size_t)blockDim.x + threadIdx.x;
  size_t stride = (size_t)gridDim.x * blockDim.x;
  for (; i < n; i += stride) p[i] = 0.0f;
}

// ---------------------------------------------------------------------------
// Flat GEMM: C[M,128] = A[M,128] @ W[128,128], W staged in LDS (64 KB).
// One wave computes one 16x16 tile; 8 waves/block cover a 16x128 strip.
// WMMA f32 16x16x4, K loop of 32 steps.
// ---------------------------------------------------------------------------
__global__ void gemm_flat_wmma(const float* __restrict__ A,
                               const float* __restrict__ W,
                               float* __restrict__ C) {
  __shared__ float Wl[128 * 128];  // 64 KB
  const int tid = threadIdx.x;     // 256 threads
  {
    const float4* W4 = (const float4*)W;
    float4* Wl4 = (float4*)Wl;
    for (int i = tid; i < 128 * 128 / 4; i += 256) Wl4[i] = W4[i];
  }
  __syncthreads();

  const int wave = tid >> 5;
  const int lane = tid & 31;
  const int row0 = blockIdx.x * 16;
  const int col0 = wave * 16;
  const int m = row0 + (lane & 15);
  const int kh = (lane >> 4) * 2;   // ISA 7.12.2: lanes16-31 hold K+2,K+3
  const int n = col0 + (lane & 15);

  v8f acc = {};
  const float* Arow = A + (size_t)m * 128;
#pragma unroll 4
  for (int kk = 0; kk < 128; kk += 4) {
    v2f a, b;
    a.x = Arow[kk + kh];
    a.y = Arow[kk + kh + 1];
    b.x = Wl[(kk + kh) * 128 + n];
    b.y = Wl[(kk + kh + 1) * 128 + n];
    acc = __builtin_amdgcn_wmma_f32_16x16x4_f32(false, a, false, b, (short)0,
                                                acc, false, false);
  }
  // C/D layout: VGPR e -> row (e) lanes0-15, row (e+8) lanes16-31; col=lane%16
  const int rbase = row0 + (lane >> 4) * 8;
#pragma unroll
  for (int e = 0; e < 8; ++e)
    C[(size_t)(rbase + e) * 128 + n] = acc[e];
}

// ---------------------------------------------------------------------------
// Batched normalized-adjacency GEMM:
//   Out[b,i,f] = relu?( dinv[b,i] * sum_j (A[b,i,j] + self*(i==j)) * dinv[b,j]
//                       * X[b,j,f]  + bias[f] )
// d = 128 fixed; one wave = one 16x16 tile; 8 waves cover all 8 f-tiles.
// grid = (k/16, B)
// ---------------------------------------------------------------------------
__global__ void gemm_dense_norm_wmma(const float* __restrict__ A,
                                     const float* __restrict__ X,
                                     const float* __restrict__ dinv,
                                     const float* __restrict__ bias,
                                     float* __restrict__ Out, int k,
                                     int addSelf, int doRelu) {
  const int b = blockIdx.y;
  const int i0 = blockIdx.x * 16;
  const int wave = threadIdx.x >> 5;
  const int lane = threadIdx.x & 31;
  const int f = wave * 16 + (lane & 15);
  const int kh = (lane >> 4) * 2;
  const int i = i0 + (lane & 15);

  const float* Ab = A + (size_t)b * k * k;
  const float* Xb = X + (size_t)b * k * 128;
  const float* dv = dinv + (size_t)b * k;
  const float* Arow = Ab + (size_t)i * k;

  v8f acc = {};
  for (int kk = 0; kk < k; kk += 4) {
    const int j0 = kk + kh;
    float a0 = Arow[j0];
    float a1 = Arow[j0 + 1];
    if (addSelf) {
      if (j0 == i) a0 += 1.0f;
      if (j0 + 1 == i) a1 += 1.0f;
    }
    v2f a, bb;
    a.x = a0 * dv[j0];
    a.y = a1 * dv[j0 + 1];
    bb.x = Xb[(size_t)j0 * 128 + f];
    bb.y = Xb[(size_t)(j0 + 1) * 128 + f];
    acc = __builtin_amdgcn_wmma_f32_16x16x4_f32(false, a, false, bb, (short)0,
                                                acc, false, false);
  }
  const int rbase = i0 + (lane >> 4) * 8;
  const float bv = bias ? bias[f] : 0.0f;
  float* Ob = Out + (size_t)b * k * 128;
#pragma unroll
  for (int e = 0; e < 8; ++e) {
    const int r = rbase + e;
    float v = acc[e] * dv[r] + bv;  // fold row-side dinv into epilogue
    if (doRelu) v = fmaxf(v, 0.0f);
    Ob[(size_t)r * 128 + f] = v;
  }
}

// ---------------------------------------------------------------------------
// Edge-list kernels
// ---------------------------------------------------------------------------
__global__ void edge_degree(const int* __restrict__ dst,
                            float* __restrict__ deg, int E) {
  int e = blockIdx.x * blockDim.x + threadIdx.x;
  if (e < E) atomicAdd(&deg[dst[e]], 1.0f);
}

__global__ void make_dinv(const float* __restrict__ deg,
                          float* __restrict__ dinv1,
                          float* __restrict__ dinv0, int n) {
  int i = blockIdx.x * blockDim.x + threadIdx.x;
  if (i >= n) return;
  float d = deg[i];
  dinv1[i] = rsqrtf(d + 1.0f);                               // GCN: deg+1
  dinv0[i] = d > 0.0f ? rsqrtf(fmaxf(d, 1.0f)) : 0.0f;       // info score
}

// one wave per edge: Acc[dst,:] += H[src,:] * dinv[src]*dinv[dst]
__global__ void edge_scatter(const int* __restrict__ src,
                             const int* __restrict__ dst,
                             const float* __restrict__ dinv,
                             const float* __restrict__ H,
                             float* __restrict__ Acc, int E) {
  int w = (int)((blockIdx.x * (size_t)blockDim.x + threadIdx.x) >> 5);
  int lane = threadIdx.x & 31;
  if (w >= E) return;
  int s = src[w], d = dst[w];
  float coef = dinv[s] * dinv[d];
  float4 v = ((const float4*)(H + (size_t)s * 128))[lane];
  float* out = Acc + (size_t)d * 128 + lane * 4;
  atomicAdd(out + 0, v.x * coef);
  atomicAdd(out + 1, v.y * coef);
  atomicAdd(out + 2, v.z * coef);
  atomicAdd(out + 3, v.w * coef);
}

__global__ void conv1_finalize(const float* __restrict__ agg,
                               const float* __restrict__ hpre,
                               const float* __restrict__ dinv1,
                               const float* __restrict__ bias,
                               float* __restrict__ H, size_t total) {
  size_t t = blockIdx.x * (size_t)blockDim.x + threadIdx.x;
  if (t >= total) return;
  int i = (int)(t >> 7), f = (int)(t & 127);
  float dv = dinv1[i];
  H[t] = fmaxf(agg[t] + hpre[t] * dv * dv + bias[f], 0.0f);
}

// s[i] = sum_f |X[i,f] - S[i,f]| ; one wave per row (128 feats)
__global__ void abs_diff_rowsum(const float* __restrict__ X,
                                const float* __restrict__ S,
                                float* __restrict__ out, int rows) {
  int w = (int)((blockIdx.x * (size_t)blockDim.x + threadIdx.x) >> 5);
  int lane = threadIdx.x & 31;
  if (w >= rows) return;
  float4 a = ((const float4*)(X + (size_t)w * 128))[lane];
  float4 b = ((const float4*)(S + (size_t)w * 128))[lane];
  float s = fabsf(a.x - b.x) + fabsf(a.y - b.y) + fabsf(a.z - b.z) +
            fabsf(a.w - b.w);
  for (int off = 16; off; off >>= 1) s += __shfl_down(s, off, 32);
  if (lane == 0) out[w] = s;
}

// ---------------------------------------------------------------------------
// Top-k per graph (rank by comparison count; ties broken by index -> unique
// ranks, matches lax.top_k descending order; downstream output is order-
// invariant anyway). One block per graph, blockDim = n (<=512).
// ---------------------------------------------------------------------------
__global__ void topk_sel(const float* __restrict__ scores, int n, int k,
                         int* __restrict__ idx, int* __restrict__ rank) {
  __shared__ float sh[512];
  int b = blockIdx.x, t = threadIdx.x;
  sh[t] = scores[(size_t)b * n + t];
  __syncthreads();
  float mine = sh[t];
  int r = 0;
  for (int u = 0; u < n; ++u) {
    float su = sh[u];
    r += (su > mine) || (su == mine && u < t);
  }
  if (r < k) idx[(size_t)b * k + r] = t;
  if (rank) rank[(size_t)b * n + t] = (r < k) ? r : -1;
}

// Out[b,r,:] = X[b*srcN + idx[b*k+r], :]
__global__ void gather_rows(const float* __restrict__ X,
                            const int* __restrict__ idx,
                            float* __restrict__ Out, int srcN, int k,
                            size_t total) {
  size_t t = blockIdx.x * (size_t)blockDim.x + threadIdx.x;
  if (t >= total) return;
  int f = (int)(t & 127);
  size_t br = t >> 7;           // b*k + r
  size_t b = br / k;
  int row = idx[br];
  Out[t] = X[((size_t)b * srcN + row) * 128 + f];
}

// Pooled adjacency from edge list: A[b, rank[src], rank[dst]] += 1
__global__ void build_adj(const int* __restrict__ src,
                          const int* __restrict__ dst,
                          const int* __restrict__ rank, float* __restrict__ A,
                          int N, int k, int E) {
  int e = blockIdx.x * blockDim.x + threadIdx.x;
  if (e >= E) return;
  int s = src[e], d = dst[e];
  int u = rank[s], v = rank[d];
  if (u >= 0 && v >= 0) {
    int b = s / N;
    atomicAdd(&A[((size_t)b * k + u) * k + v], 1.0f);
  }
}

// Out[b,r,c] = A[b, idx[b*k+r], idx[b*k+c]]  (A is [B,n,n])
__global__ void gather_adj(const float* __restrict__ A,
                           const int* __restrict__ idx, float* __restrict__ Out,
                           int n, int k, size_t total) {
  size_t t = blockIdx.x * (size_t)blockDim.x + threadIdx.x;
  if (t >= total) return;
  int c = (int)(t % k);
  size_t br = t / k;            // b*k + r
  size_t b = br / k;
  int ir = idx[br];
  int ic = idx[b * k + c];
  Out[t] = A[((size_t)b * n + ir) * n + ic];
}

// q[b,i] = hk[b,i,:] . att[:128] ; s[b,i] = hk[b,i,:] . att[128:]
__global__ void attn_qs(const float* __restrict__ Xk,
                        const float* __restrict__ att, float* __restrict__ q,
                        float* __restrict__ s, int rows) {
  int w = (int)((blockIdx.x * (size_t)blockDim.x + threadIdx.x) >> 5);
  int lane = threadIdx.x & 31;
  if (w >= rows) return;
  float4 xv = ((const float4*)(Xk + (size_t)w * 128))[lane];
  float4 av = ((const float4*)att)[lane];
  float4 bv = ((const float4*)(att + 128))[lane];
  float qd = xv.x * av.x + xv.y * av.y + xv.z * av.z + xv.w * av.w;
  float sd = xv.x * bv.x + xv.y * bv.y + xv.z * bv.z + xv.w * bv.w;
  for (int off = 16; off; off >>= 1) {
    qd += __shfl_down(qd, off, 32);
    sd += __shfl_down(sd, off, 32);
  }
  if (lane == 0) {
    q[w] = qd;
    s[w] = sd;
  }
}

// A[b,i,:] = softmax_j(leaky(q_i + s_j)) + lamb * A[b,i,:]   (in place)
// one wave per row; k in {128, 256}
__global__ void structure_softmax(const float* __restrict__ q,
                                  const float* __restrict__ s,
                                  float* __restrict__ A, int k, float lamb,
                                  float slope, int rows) {
  int w = (int)((blockIdx.x * (size_t)blockDim.x + threadIdx.x) >> 5);
  int lane = threadIdx.x & 31;
  if (w >= rows) return;
  int b = w / k;
  const float* srow = s + (size_t)b * k;
  float qi = q[w];
  float vals[8];
  const int cnt = k >> 5;
  float m = -3.4e38f;
  for (int t = 0; t < cnt; ++t) {
    float z = qi + srow[lane + t * 32];
    z = z > 0.0f ? z : z * slope;
    vals[t] = z;
    m = fmaxf(m, z);
  }
  for (int off = 16; off; off >>= 1) m = fmaxf(m, __shfl_xor(m, off, 32));
  float sum = 0.0f;
  for (int t = 0; t < cnt; ++t) {
    vals[t] = __expf(vals[t] - m);
    sum += vals[t];
  }
  for (int off = 16; off; off >>= 1) sum += __shfl_xor(sum, off, 32);
  float inv = 1.0f / sum;
  float* Arow = A + (size_t)w * k;
  for (int t = 0; t < cnt; ++t) {
    int j = lane + t * 32;
    Arow[j] = vals[t] * inv + lamb * Arow[j];
  }
}

// rowsum of A[b,i,:] -> conv dinv (with +I) and info dinv (without)
__global__ void adj_rowsum(const float* __restrict__ A,
                           float* __restrict__ dconv,
                           float* __restrict__ dinfo, int k, int rows) {
  int w = (int)((blockIdx.x * (size_t)blockDim.x + threadIdx.x) >> 5);
  int lane = threadIdx.x & 31;
  if (w >= rows) return;
  const float* Arow = A + (size_t)w * k;
  float sum = 0.0f;
  for (int j = lane; j < k; j += 32) sum += Arow[j];
  for (int off = 16; off; off >>= 1) sum += __shfl_xor(sum, off, 32);
  if (lane == 0) {
    if (dconv) dconv[w] = rsqrtf(fmaxf(sum + 1.0f, 1e-12f));
    if (dinfo) dinfo[w] = sum > 0.0f ? rsqrtf(fmaxf(sum, 1e-12f)) : 0.0f;
  }
}

// out[b, 0:128] (+)= relu(max_i X[b,i,f]); out[b,128:256] (+)= relu(mean_i)
__global__ void readout_acc(const float* __restrict__ X, float* __restrict__ out,
                            int k, int accumulate, int B) {
  int t = blockIdx.x * blockDim.x + threadIdx.x;
  if (t >= B * 256) return;
  int b = t >> 8, f2 = t & 255;
  const float* Xb = X + (size_t)b * k * 128;
  float v;
  if (f2 < 128) {
    v = -3.4e38f;
    for (int i = 0; i < k; ++i) v = fmaxf(v, Xb[(size_t)i * 128 + f2]);
  } else {
    int f = f2 - 128;
    float s = 0.0f;
    for (int i = 0; i < k; ++i) s += Xb[(size_t)i * 128 + f];
    v = s / (float)k;
  }
  v = fmaxf(v, 0.0f);
  if (accumulate)
    out[t] += v;
  else
    out[t] = v;
}

// ---------------------------------------------------------------------------
// Host orchestration
// ---------------------------------------------------------------------------
extern "C" void kernel_launch(void* const* d_in, const int* in_sizes, int n_in,
                              void* d_out, int out_size, void* d_ws,
                              size_t ws_size, hipStream_t stream) {
  const float* x = (const float*)d_in[0];
  const float* W1 = (const float*)d_in[1];
  const float* b1 = (const float*)d_in[2];
  const float* W2 = (const float*)d_in[3];
  const float* b2 = (const float*)d_in[4];
  const float* W3 = (const float*)d_in[5];
  const float* b3 = (const float*)d_in[6];
  const float* att1 = (const float*)d_in[7];
  const float* att2 = (const float*)d_in[8];
  const int* esrc = (const int*)d_in[9];
  const int* edst = (const int*)d_in[10];
  float* out = (float*)d_out;

  const int dF = 128;
  const int NT = in_sizes[0] / dF;  // 131072 nodes total
  const int E = in_sizes[9];        // 2,097,152 edges
  const int B = 256;                // num_graphs (setup constant)
  const int N = NT / B;             // 512
  const int k1 = N / 2;             // 256
  const int k2 = N / 4;             // 128
  const size_t NTd = (size_t)NT * dF;

  // Workspace slabs (float), with lifetime-based reuse:
  float* ws = (float*)d_ws;
  float* S0 = ws;              // h1pre -> smooth1 -> hw2|h2 -> hw3|h3
  float* S1 = ws + NTd;        // agg -> A1 (B*k1*k1 == NTd)
  float* S2 = ws + 2 * NTd;    // h -> smooth2 | h2k
  float* S3 = ws + 3 * NTd;    // hk -> A1k/A2
  float* deg = ws + 3 * NTd + (size_t)B * k1 * dF;
  float* dinv1 = deg + NT;
  float* dinv0 = dinv1 + NT;
  float* s1 = dinv0 + NT;
  float* q1 = s1 + NT;
  float* sq1 = q1 + (size_t)B * k1;
  float* dA = sq1 + (size_t)B * k1;
  float* dAi = dA + (size_t)B * k1;
  float* s2 = dAi + (size_t)B * k1;
  float* q2 = s2 + (size_t)B * k1;
  float* sq2 = q2 + (size_t)B * k2;
  float* dA2 = sq2 + (size_t)B * k2;
  int* rank = (int*)(dA2 + (size_t)B * k2);
  int* idx1 = rank + NT;
  int* idx2 = idx1 + (size_t)B * k1;

  float* h1pre = S0;
  float* agg = S1;
  float* h = S2;
  float* smooth1 = S0;
  float* hk = S3;
  float* A1 = S1;
  float* hw2 = S0;
  float* h2 = S0 + (size_t)B * k1 * dF;
  float* smooth2 = S2;
  float* h2k = S2 + (size_t)B * k1 * dF;
  float* A1k = S3;  // becomes A2 in place
  float* hw3 = S0;
  float* h3 = S0 + (size_t)B * k2 * dF;

  const float LAMB = 1.0f, SLOPE = 0.2f;
  const int ZB = 4096;

  // ---- conv1 (edge GCN) ----
  zero_f32<<<ZB, 256, 0, stream>>>(deg, (size_t)NT);
  edge_degree<<<CDIV(E, 256), 256, 0, stream>>>(edst, deg, E);
  make_dinv<<<CDIV(NT, 256), 256, 0, stream>>>(deg, dinv1, dinv0, NT);
  gemm_flat_wmma<<<NT / 16, 256, 0, stream>>>(x, W1, h1pre);
  zero_f32<<<ZB, 256, 0, stream>>>(agg, NTd);
  edge_scatter<<<CDIV((size_t)E * 32, 256), 256, 0, stream>>>(esrc, edst, dinv1,
                                                              h1pre, agg, E);
  conv1_finalize<<<CDIV(NTd, 256), 256, 0, stream>>>(agg, h1pre, dinv1, b1, h,
                                                     NTd);
  // ---- info score 1 ----
  zero_f32<<<ZB, 256, 0, stream>>>(smooth1, NTd);
  edge_scatter<<<CDIV((size_t)E * 32, 256), 256, 0, stream>>>(esrc, edst, dinv0,
                                                              h, smooth1, E);
  abs_diff_rowsum<<<CDIV((size_t)NT * 32, 256), 256, 0, stream>>>(h, smooth1,
                                                                  s1, NT);
  // ---- pool1 ----
  topk_sel<<<B, N, 0, stream>>>(s1, N, k1, idx1, rank);
  gather_rows<<<CDIV((size_t)B * k1 * dF, 256), 256, 0, stream>>>(
      h, idx1, hk, N, k1, (size_t)B * k1 * dF);
  zero_f32<<<ZB, 256, 0, stream>>>(A1, (size_t)B * k1 * k1);
  build_adj<<<CDIV(E, 256), 256, 0, stream>>>(esrc, edst, rank, A1, N, k1, E);
  attn_qs<<<CDIV((size_t)B * k1 * 32, 256), 256, 0, stream>>>(hk, att1, q1, sq1,
                                                              B * k1);
  structure_softmax<<<CDIV((size_t)B * k1 * 32, 256), 256, 0, stream>>>(
      q1, sq1, A1, k1, LAMB, SLOPE, B * k1);
  readout_acc<<<CDIV(B * 256, 256), 256, 0, stream>>>(hk, out, k1, 0, B);

  // ---- conv2 (dense GCN) ----
  gemm_flat_wmma<<<(B * k1) / 16, 256, 0, stream>>>(hk, W2, hw2);
  adj_rowsum<<<CDIV((size_t)B * k1 * 32, 256), 256, 0, stream>>>(A1, dA, dAi,
                                                                 k1, B * k1);
  gemm_dense_norm_wmma<<<dim3(k1 / 16, B), 256, 0, stream>>>(A1, hw2, dA, b2,
                                                             h2, k1, 1, 1);
  // ---- info score 2 ----
  gemm_dense_norm_wmma<<<dim3(k1 / 16, B), 256, 0, stream>>>(
      A1, h2, dAi, (const float*)nullptr, smooth2, k1, 0, 0);
  abs_diff_rowsum<<<CDIV((size_t)B * k1 * 32, 256), 256, 0, stream>>>(
      h2, smooth2, s2, B * k1);
  // ---- pool2 ----
  topk_sel<<<B, k1, 0, stream>>>(s2, k1, k2, idx2, (int*)nullptr);
  gather_rows<<<CDIV((size_t)B * k2 * dF, 256), 256, 0, stream>>>(
      h2, idx2, h2k, k1, k2, (size_t)B * k2 * dF);
  gather_adj<<<CDIV((size_t)B * k2 * k2, 256), 256, 0, stream>>>(
      A1, idx2, A1k, k1, k2, (size_t)B * k2 * k2);
  attn_qs<<<CDIV((size_t)B * k2 * 32, 256), 256, 0, stream>>>(h2k, att2, q2,
                                                              sq2, B * k2);
  structure_softmax<<<CDIV((size_t)B * k2 * 32, 256), 256, 0, stream>>>(
      q2, sq2, A1k, k2, LAMB, SLOPE, B * k2);
  readout_acc<<<CDIV(B * 256, 256), 256, 0, stream>>>(h2k, out, k2, 1, B);

  // ---- conv3 ----
  gemm_flat_wmma<<<(B * k2) / 16, 256, 0, stream>>>(h2k, W3, hw3);
  adj_rowsum<<<CDIV((size_t)B * k2 * 32, 256), 256, 0, stream>>>(
      A1k, dA2, (float*)nullptr, k2, B * k2);
  gemm_dense_norm_wmma<<<dim3(k2 / 16, B), 256, 0, stream>>>(A1k, hw3, dA2, b3,
                                                             h3, k2, 1, 1);
  readout_acc<<<CDIV(B * 256, 256), 256, 0, stream>>>(h3, out, k2, 1, B);
}